// VectorQuantizer_41910290874589
// MI455X (gfx1250) — compile-verified
//
#include <hip/hip_runtime.h>
#include <hip/hip_bf16.h>
#include <math.h>

typedef __attribute__((ext_vector_type(2))) float v2f;
typedef __attribute__((ext_vector_type(8))) float v8f;
typedef __attribute__((ext_vector_type(4))) unsigned int v4u;
typedef __attribute__((ext_vector_type(8))) int v8i_t;
typedef __attribute__((ext_vector_type(4))) int v4i_t;

#define E_DIM   64
#define NUM_E   512
#define NROWS   131072                 // 32*64*64
#define XQ_ELEMS 8388608               // 32*64*64*64
#define LOSS_OFF 0
#define XQ_OFF   1
#define PERP_OFF (1 + XQ_ELEMS)        // 8388609
#define ENC_OFF  (2 + XQ_ELEMS)        // 8388610

#define CB_STRIDE  68                  // 64 data + 4 pad floats -> conflict-free ds_load_b64
#define CBP_FLOATS (NUM_E * CB_STRIDE) // 34816 floats = 139,264 B
#define CN2_OFF_F  CBP_FLOATS
#define IDX_OFF_F  (CBP_FLOATS + NUM_E)
#define SMEM_FLOATS (CBP_FLOATS + NUM_E + 64)

__global__ void vq_init(float* out, unsigned int* hist) {
    int tid = threadIdx.x;
    hist[tid] = 0u;
    if (tid == 0) out[LOSS_OFF] = 0.0f;
}

__global__ void vq_main(const float* __restrict__ x,
                        const float* __restrict__ cb,
                        float* __restrict__ out,
                        unsigned int* __restrict__ hist) {
    extern __shared__ float smem[];
    int* idx_smem = (int*)(smem + IDX_OFF_F);
    const int tid = threadIdx.x;

    // ---- TDM: stage codebook [512 x 64] f32 into LDS with 4-DWORD row pad ----
    // D# group0: count=1, lds_addr, 57b global_addr, type=2 ("image")
    // D# group1: data_size=4B(2), pad_enable, pad_interval=64 DW (enc 5),
    //            pad_amount=4 DW (enc 3), tensor 64x512, tile 64x512, stride0=64
    if (tid < 32) {   // wave 0 issues the tensor DMA
        const unsigned long long gaddr = (unsigned long long)(uintptr_t)cb;
        const unsigned int lds_off = (unsigned int)(uintptr_t)smem;  // LDS byte offset
        v4u g0;
        g0[0] = 1u;                                    // count=1, user mode
        g0[1] = lds_off;                               // lds_addr (bytes)
        g0[2] = (unsigned int)(gaddr & 0xFFFFFFFFu);   // global_addr[31:0]
        g0[3] = (unsigned int)((gaddr >> 32) & 0x1FFFFFFu) | (2u << 30); // [56:32] | type=2
        v8i_t g1;
        g1[0] = (2 << 16) | (1 << 20) | (5 << 22) | (3 << 25); // dsz=4B, pad 4DW every 64DW
        g1[1] = (int)(64u << 16);      // tensor_dim0[15:0] @ bits63:48
        g1[2] = (int)(512u << 16);     // tensor_dim1[15:0] @ bits95:80
        g1[3] = (int)(64u << 16);      // tile_dim0 @ bits127:112
        g1[4] = 512;                   // tile_dim1 @ bits143:128
        g1[5] = 64;                    // tensor_dim0_stride[31:0] @ bits191:160
        g1[6] = 0;
        g1[7] = 0;
        v4i_t gz4 = {0, 0, 0, 0};              // groups 2/3 unused (2D tensor)
        v8i_t gz8 = {0, 0, 0, 0, 0, 0, 0, 0};  // extra group (clang-23 6-arg form)
        __builtin_amdgcn_tensor_load_to_lds(g0, g1, gz4, gz4, gz8, 0);
    }

    // ---- ||c||^2 per entry from global (overlaps with the TDM transfer) ----
    for (int e = tid; e < NUM_E; e += 128) {
        const float4* rp = (const float4*)(cb + e * E_DIM);
        float s = 0.f;
        #pragma unroll
        for (int q = 0; q < 16; ++q) {
            const float4 vv = rp[q];
            s += vv.x * vv.x + vv.y * vv.y + vv.z * vv.z + vv.w * vv.w;
        }
        smem[CN2_OFF_F + e] = s;
    }
    if (tid < 32) __builtin_amdgcn_s_wait_tensorcnt(0);
    __syncthreads();

    const int wave  = tid >> 5;
    const int lane  = tid & 31;
    const int m     = lane & 15;          // row within M-tile / N column within tile
    const int khalf = (lane >> 4) << 1;   // 0 or 2 (ISA fp32 A/B fragment layout)
    const int nw    = blockIdx.x * 64 + wave * 16;   // first flat row of this wave
    const int bb = nw >> 12, hh = (nw >> 6) & 63, w0 = nw & 63;
    const int x_base = bb * 262144 + hh * 64 + w0;   // NCHW, feature stride 4096

    // ---- A fragments: 16 K-chunks of 16x4 fp32 ----
    v2f a_frag[16];
    #pragma unroll
    for (int kc = 0; kc < 16; ++kc) {
        const int k = kc * 4 + khalf;
        a_frag[kc][0] = x[x_base + m + k * 4096];
        a_frag[kc][1] = x[x_base + m + (k + 1) * 4096];
    }

    float best_val[8];
    int   best_idx[8];
    #pragma unroll
    for (int v = 0; v < 8; ++v) { best_val[v] = 3.0e38f; best_idx[v] = 0; }

    // ---- 32 N-tiles; 16 v_wmma_f32_16x16x4_f32 each, two independent chains ----
    for (int t = 0; t < 32; ++t) {
        const int ncol = t * 16 + m;
        const float* bbase = smem + ncol * CB_STRIDE; // conflict-free b64 at imm offsets
        v8f acc0 = {0.f, 0.f, 0.f, 0.f, 0.f, 0.f, 0.f, 0.f};
        v8f acc1 = {0.f, 0.f, 0.f, 0.f, 0.f, 0.f, 0.f, 0.f};
        #pragma unroll
        for (int kc = 0; kc < 16; kc += 2) {
            const v2f b0 = *(const v2f*)(bbase + kc * 4 + khalf);
            const v2f b1 = *(const v2f*)(bbase + (kc + 1) * 4 + khalf);
            acc0 = __builtin_amdgcn_wmma_f32_16x16x4_f32(
                false, a_frag[kc], false, b0, (short)0, acc0, false, false);
            acc1 = __builtin_amdgcn_wmma_f32_16x16x4_f32(
                false, a_frag[kc + 1], false, b1, (short)0, acc1, false, false);
        }
        const float cn2v = smem[CN2_OFF_F + ncol];
        #pragma unroll
        for (int v = 0; v < 8; ++v) {
            const float s = fmaf(-2.0f, acc0[v] + acc1[v], cn2v); // ||c||^2 - 2 x.c
            if (s < best_val[v]) { best_val[v] = s; best_idx[v] = ncol; }
        }
    }

    // ---- argmin reduce across the 16 lanes of each N-group (wave32 shuffles) ----
    #pragma unroll
    for (int mask = 8; mask >= 1; mask >>= 1) {
        #pragma unroll
        for (int v = 0; v < 8; ++v) {
            const float ov = __shfl_xor(best_val[v], mask, 32);
            const int   oi = __shfl_xor(best_idx[v], mask, 32);
            if (ov < best_val[v] || (ov == best_val[v] && oi < best_idx[v])) {
                best_val[v] = ov; best_idx[v] = oi;
            }
        }
    }
    if (m == 0) {
        const int g = lane >> 4;
        #pragma unroll
        for (int v = 0; v < 8; ++v) {
            const int row = v + 8 * g;          // C layout: VGPR v -> M = v + 8*(lane>=16)
            idx_smem[wave * 16 + row] = best_idx[v];
            atomicAdd(&hist[best_idx[v]], 1u);
        }
    }
    __syncthreads();

    // ---- x_q_st (NCHW, coalesced) + loss partial ----
    const int myidx = idx_smem[wave * 16 + m];
    const float* qrow = smem + myidx * CB_STRIDE;
    const int cgrp  = lane >> 4;
    float lsum = 0.f;
    #pragma unroll 4
    for (int cc = 0; cc < 32; ++cc) {
        const int   c  = 2 * cc + cgrp;
        const float q  = qrow[c];
        const int   off = x_base + m + c * 4096;
        const float xv = x[off];
        out[XQ_OFF + off] = xv + (q - xv);     // straight-through, match jnp rounding
        const float d = q - xv;
        lsum += d * d;
    }
    #pragma unroll
    for (int mask = 16; mask >= 1; mask >>= 1) lsum += __shfl_xor(lsum, mask, 32);
    if (lane == 0) atomicAdd(&out[LOSS_OFF], lsum * (1.25f / 8388608.0f));

    // ---- one-hot encodings: 16 rows x 512, aligned b64 stores ----
    float* encp = out + (size_t)ENC_OFF + (size_t)nw * NUM_E;
    for (int j = 0; j < 16; ++j) {
        const int bidx = idx_smem[wave * 16 + j];
        float* rowp = encp + j * NUM_E;
        #pragma unroll
        for (int q = 0; q < 8; ++q) {
            const int e0 = q * 64 + lane * 2;
            float2 vv;
            vv.x = (e0     == bidx) ? 1.0f : 0.0f;
            vv.y = (e0 + 1 == bidx) ? 1.0f : 0.0f;
            *(float2*)(rowp + e0) = vv;
        }
    }
}

__global__ void vq_perplexity(const unsigned int* __restrict__ hist,
                              float* __restrict__ out) {
    __shared__ float red[16];
    const int tid = threadIdx.x;
    const int lane = tid & 31;
    const int wave = tid >> 5;
    const float p = (float)hist[tid] * (1.0f / (float)NROWS);
    float term = p * logf(p + 1e-10f);
    #pragma unroll
    for (int mask = 16; mask >= 1; mask >>= 1) term += __shfl_xor(term, mask, 32);
    if (lane == 0) red[wave] = term;
    __syncthreads();
    if (tid < 16) {
        float s = red[tid];
        #pragma unroll
        for (int mask = 8; mask >= 1; mask >>= 1) s += __shfl_xor(s, mask, 32);
        if (tid == 0) out[PERP_OFF] = expf(-s);
    }
}

extern "C" void kernel_launch(void* const* d_in, const int* in_sizes, int n_in,
                              void* d_out, int out_size, void* d_ws, size_t ws_size,
                              hipStream_t stream) {
    const float* x  = (const float*)d_in[0];   // [32,64,64,64] fp32 NCHW
    const float* cb = (const float*)d_in[1];   // [512,64] fp32
    float* out = (float*)d_out;
    unsigned int* hist = (unsigned int*)d_ws;  // 512 counters

    vq_init<<<dim3(1), dim3(512), 0, stream>>>(out, hist);
    const size_t smem_bytes = (size_t)SMEM_FLOATS * sizeof(float);
    vq_main<<<dim3(NROWS / 64), dim3(128), smem_bytes, stream>>>(x, cb, out, hist);
    vq_perplexity<<<dim3(1), dim3(512), 0, stream>>>(hist, out);
}